// SkipGRU_87050397155383
// MI455X (gfx1250) — compile-verified
//
#include <hip/hip_runtime.h>
#include <cstddef>

// ---------------------------------------------------------------------------
// Skip-GRU, MI455X (gfx1250, wave32, WMMA).
//
// Only the lane-(P-1) skip chain contributes to outputs[:, -1, :]  =>  run the
// recurrence for that single chain (24x dead-code elimination).
//
// Two-phase schedule:
//   Phase 1 (parallel):   MX[c] = x[:, t_c, :] @ Wk + bi   for all 8 chunks
//                         ([2048,512]x[512,1536] fp32 WMMA, 4096 waves)
//   Phase 2 (sequential): per chunk, MI = h @ Wr, gates, blend.
//                         Split-K x4 inside a 128-thread block (LDS reduce)
//                         to shorten the recurrent critical path and give
//                         4x the resident waves for latency hiding.
// All matrix math uses V_WMMA_F32_16X16X4_F32 (exact fp32 path).
// ---------------------------------------------------------------------------

typedef float v2f __attribute__((ext_vector_type(2)));
typedef float v8f __attribute__((ext_vector_type(8)));

#define BB     256
#define TT     192
#define FF     512
#define UU     512
#define PSKIP  24
#define NCHUNK 8
#define N3     (3 * UU)   // 1536

__global__ __launch_bounds__(32)
void skipgru_zero_kernel(float* __restrict__ p, int n) {
    int i = blockIdx.x * 32 + threadIdx.x;
    if (i < n) p[i] = 0.0f;
}

// ---------------------------------------------------------------------------
// Phase 1: MX[c][b][g*U+n] = sum_k x[b, t_c, k] * Wk[k, g*U+n] + bi[g*U+n]
// One wave per (16x16 tile of [B rows x U cols]) x chunk; 3 gate columns each.
// ---------------------------------------------------------------------------
__global__ __launch_bounds__(32)
void skipgru_mx_kernel(const float* __restrict__ x,   // [B,T,F]
                       const float* __restrict__ Wk,  // [F,3U]
                       const float* __restrict__ bi,  // [3U]
                       float* __restrict__ mx)        // [NC,B,3U]
{
    const int lane = threadIdx.x;
    const int half = lane >> 4;
    const int lm   = lane & 15;
    const int m0   = blockIdx.y * 16;
    const int n0   = blockIdx.x * 16;
    const int c    = blockIdx.z;
    const int t    = c * PSKIP + (PSKIP - 1);

    const int    rowA = m0 + lm;
    const float* xrow = x + ((size_t)rowA * TT + t) * FF;
    const int    colB = n0 + lm;

    v8f acc_z = {}, acc_r = {}, acc_h = {};

    for (int k0 = 0; k0 < FF; k0 += 4) {
        const int ka = k0 + 2 * half;
        v2f a = *(const v2f*)(xrow + ka);

        const float* wk = Wk + (size_t)ka * N3 + colB;
        v2f bz = { wk[0],      wk[N3]          };
        v2f brv= { wk[UU],     wk[N3 + UU]     };
        v2f bh = { wk[2 * UU], wk[N3 + 2 * UU] };

        acc_z = __builtin_amdgcn_wmma_f32_16x16x4_f32(false, a, false, bz,
                                                      (short)0, acc_z, false, false);
        acc_r = __builtin_amdgcn_wmma_f32_16x16x4_f32(false, a, false, brv,
                                                      (short)0, acc_r, false, false);
        acc_h = __builtin_amdgcn_wmma_f32_16x16x4_f32(false, a, false, bh,
                                                      (short)0, acc_h, false, false);
    }

    const float biz = bi[n0 + lm];
    const float bir = bi[UU + n0 + lm];
    const float bih = bi[2 * UU + n0 + lm];

#pragma unroll
    for (int v = 0; v < 8; ++v) {
        const int    row  = m0 + v + 8 * half;
        const size_t base = ((size_t)c * BB + row) * N3 + (n0 + lm);
        mx[base]          = acc_z[v] + biz;
        mx[base + UU]     = acc_r[v] + bir;
        mx[base + 2 * UU] = acc_h[v] + bih;
    }
}

// ---------------------------------------------------------------------------
// Phase 2: one 128-thread block (4 waves) per 16x16 tile of h_new.
// Wave w accumulates MI over K in [w*128, (w+1)*128); LDS reduce; wave 0
// applies gates using precomputed MX and blends with h_old.
// ---------------------------------------------------------------------------
__global__ __launch_bounds__(128)
void skipgru_step_kernel(const float* __restrict__ mxc,  // [B,3U] chunk slice
                         const float* __restrict__ Wr,   // [U,3U]
                         const float* __restrict__ br,   // [3U]
                         const float* __restrict__ h_in, // [B,U]
                         float* __restrict__ h_out,      // [B,U]
                         float* __restrict__ out)        // [B,U] or null
{
    __shared__ float red[3][4][32 * 8];   // 12 KB: acc x wave x (lane,v)

    const int lane = threadIdx.x & 31;
    const int w    = threadIdx.x >> 5;    // 0..3 : K-chunk owner
    const int half = lane >> 4;
    const int lm   = lane & 15;
    const int m0   = blockIdx.y * 16;
    const int n0   = blockIdx.x * 16;

    const int    rowA = m0 + lm;
    const float* hrow = h_in + (size_t)rowA * UU;
    const int    colB = n0 + lm;

    v8f acc_z = {}, acc_r = {}, acc_h = {};

    const int kbeg = w * (UU / 4);
    const int kend = kbeg + (UU / 4);
    for (int k0 = kbeg; k0 < kend; k0 += 4) {
        const int ka = k0 + 2 * half;
        v2f a = *(const v2f*)(hrow + ka);

        const float* wr = Wr + (size_t)ka * N3 + colB;
        v2f bz = { wr[0],      wr[N3]          };
        v2f brv= { wr[UU],     wr[N3 + UU]     };
        v2f bh = { wr[2 * UU], wr[N3 + 2 * UU] };

        acc_z = __builtin_amdgcn_wmma_f32_16x16x4_f32(false, a, false, bz,
                                                      (short)0, acc_z, false, false);
        acc_r = __builtin_amdgcn_wmma_f32_16x16x4_f32(false, a, false, brv,
                                                      (short)0, acc_r, false, false);
        acc_h = __builtin_amdgcn_wmma_f32_16x16x4_f32(false, a, false, bh,
                                                      (short)0, acc_h, false, false);
    }

    if (w != 0) {
#pragma unroll
        for (int v = 0; v < 8; ++v) {
            red[0][w][lane * 8 + v] = acc_z[v];
            red[1][w][lane * 8 + v] = acc_r[v];
            red[2][w][lane * 8 + v] = acc_h[v];
        }
    }
    __syncthreads();

    if (w == 0) {
#pragma unroll
        for (int v = 0; v < 8; ++v) {
            acc_z[v] += red[0][1][lane * 8 + v] + red[0][2][lane * 8 + v]
                      + red[0][3][lane * 8 + v];
            acc_r[v] += red[1][1][lane * 8 + v] + red[1][2][lane * 8 + v]
                      + red[1][3][lane * 8 + v];
            acc_h[v] += red[2][1][lane * 8 + v] + red[2][2][lane * 8 + v]
                      + red[2][3][lane * 8 + v];
        }

        const float brz = br[n0 + lm];
        const float brr = br[UU + n0 + lm];
        const float brh = br[2 * UU + n0 + lm];

#pragma unroll
        for (int v = 0; v < 8; ++v) {
            const int    row  = m0 + v + 8 * half;
            const size_t mbase= (size_t)row * N3 + (n0 + lm);
            const size_t hidx = (size_t)row * UU + (n0 + lm);

            const float mxz = mxc[mbase];
            const float mxr = mxc[mbase + UU];
            const float mxh = mxc[mbase + 2 * UU];

            const float z = 1.0f / (1.0f + __expf(-(mxz + acc_z[v] + brz)));
            const float r = 1.0f / (1.0f + __expf(-(mxr + acc_r[v] + brr)));
            float hh = mxh + r * (acc_h[v] + brh);
            hh = hh > 0.0f ? hh : 0.0f;

            const float hold = h_in[hidx];
            const float hn   = z * hold + (1.0f - z) * hh;
            h_out[hidx] = hn;
            if (out) out[hidx] = hn;
        }
    }
}

// ---------------------------------------------------------------------------
// Fallback (small d_ws): round-1 fused kernel — both GEMMs per step in one
// wave per tile. Needs only 1 MB of scratch.
// ---------------------------------------------------------------------------
__global__ __launch_bounds__(32)
void skipgru_fused_kernel(const float* __restrict__ x,
                          const float* __restrict__ Wk,
                          const float* __restrict__ Wr,
                          const float* __restrict__ bi,
                          const float* __restrict__ br,
                          const float* __restrict__ h_in,
                          float* __restrict__ h_out,
                          float* __restrict__ out,
                          int t)
{
    const int lane = threadIdx.x;
    const int half = lane >> 4;
    const int lm   = lane & 15;
    const int m0   = blockIdx.y * 16;
    const int n0   = blockIdx.x * 16;

    const int    rowA = m0 + lm;
    const float* xrow = x    + ((size_t)rowA * TT + t) * FF;
    const float* hrow = h_in +  (size_t)rowA * UU;
    const int    colB = n0 + lm;

    v8f acc_z = {}, acc_r = {}, acc_xh = {}, acc_rh = {};

    for (int k0 = 0; k0 < FF; k0 += 4) {
        const int ka = k0 + 2 * half;
        v2f ax = *(const v2f*)(xrow + ka);
        v2f ah = *(const v2f*)(hrow + ka);

        const float* wk = Wk + (size_t)ka * N3 + colB;
        const float* wr = Wr + (size_t)ka * N3 + colB;
        v2f bkz = { wk[0],      wk[N3]          };
        v2f bkr = { wk[UU],     wk[N3 + UU]     };
        v2f bkh = { wk[2 * UU], wk[N3 + 2 * UU] };
        v2f brz = { wr[0],      wr[N3]          };
        v2f brr = { wr[UU],     wr[N3 + UU]     };
        v2f brh = { wr[2 * UU], wr[N3 + 2 * UU] };

        acc_z  = __builtin_amdgcn_wmma_f32_16x16x4_f32(false, ax, false, bkz,
                                                       (short)0, acc_z,  false, false);
        acc_z  = __builtin_amdgcn_wmma_f32_16x16x4_f32(false, ah, false, brz,
                                                       (short)0, acc_z,  false, false);
        acc_r  = __builtin_amdgcn_wmma_f32_16x16x4_f32(false, ax, false, bkr,
                                                       (short)0, acc_r,  false, false);
        acc_r  = __builtin_amdgcn_wmma_f32_16x16x4_f32(false, ah, false, brr,
                                                       (short)0, acc_r,  false, false);
        acc_xh = __builtin_amdgcn_wmma_f32_16x16x4_f32(false, ax, false, bkh,
                                                       (short)0, acc_xh, false, false);
        acc_rh = __builtin_amdgcn_wmma_f32_16x16x4_f32(false, ah, false, brh,
                                                       (short)0, acc_rh, false, false);
    }

    const float bz   = bi[n0 + lm]          + br[n0 + lm];
    const float brr2 = bi[UU + n0 + lm]     + br[UU + n0 + lm];
    const float bxh  = bi[2 * UU + n0 + lm];
    const float brh2 = br[2 * UU + n0 + lm];

#pragma unroll
    for (int v = 0; v < 8; ++v) {
        const int    row = m0 + v + 8 * half;
        const size_t idx = (size_t)row * UU + (n0 + lm);

        const float z  = 1.0f / (1.0f + __expf(-(acc_z[v] + bz)));
        const float r  = 1.0f / (1.0f + __expf(-(acc_r[v] + brr2)));
        float hh = acc_xh[v] + bxh + r * (acc_rh[v] + brh2);
        hh = hh > 0.0f ? hh : 0.0f;

        const float hold = h_in[idx];
        const float hn   = z * hold + (1.0f - z) * hh;
        h_out[idx] = hn;
        if (out) out[idx] = hn;
    }
}

extern "C" void kernel_launch(void* const* d_in, const int* in_sizes, int n_in,
                              void* d_out, int out_size, void* d_ws, size_t ws_size,
                              hipStream_t stream) {
    const float* x  = (const float*)d_in[0];  // [256,192,512]
    const float* Wk = (const float*)d_in[1];  // [512,1536]
    const float* Wr = (const float*)d_in[2];  // [512,1536]
    const float* bi = (const float*)d_in[3];  // [1536]
    const float* br = (const float*)d_in[4];  // [1536]
    float* out = (float*)d_out;               // [256,512]

    float* hA = (float*)d_ws;                 // h state double buffer
    float* hB = hA + (size_t)BB * UU;
    float* MX = hB + (size_t)BB * UU;         // [NC, B, 3U]

    const size_t need_two_phase =
        (2 * (size_t)BB * UU + (size_t)NCHUNK * BB * N3) * sizeof(float);

    // h0 = 0
    {
        const int n = BB * UU;
        skipgru_zero_kernel<<<(n + 31) / 32, 32, 0, stream>>>(hA, n);
    }

    if (ws_size >= need_two_phase) {
        // Phase 1: all 8 chunks' input-path GEMMs, fully parallel.
        dim3 g1(UU / 16, BB / 16, NCHUNK);     // (32, 16, 8) waves
        skipgru_mx_kernel<<<g1, 32, 0, stream>>>(x, Wk, bi, MX);

        // Phase 2: sequential recurrent GEMM + gates, split-K x4 per tile.
        dim3 g2(UU / 16, BB / 16);             // (32, 16) blocks of 4 waves
        for (int c = 0; c < NCHUNK; ++c) {
            const float* hin  = (c & 1) ? hB : hA;
            float*       hout = (c & 1) ? hA : hB;
            skipgru_step_kernel<<<g2, 128, 0, stream>>>(
                MX + (size_t)c * BB * N3, Wr, br, hin, hout,
                (c == NCHUNK - 1) ? out : nullptr);
        }
    } else {
        // Fallback: fused per-step kernel (1 MB scratch).
        dim3 grid(UU / 16, BB / 16);
        for (int c = 0; c < NCHUNK; ++c) {
            const int t = c * PSKIP + (PSKIP - 1);
            const float* hin  = (c & 1) ? hB : hA;
            float*       hout = (c & 1) ? hA : hB;
            skipgru_fused_kernel<<<grid, 32, 0, stream>>>(
                x, Wk, Wr, bi, br, hin, hout,
                (c == NCHUNK - 1) ? out : nullptr, t);
        }
    }
}